// EdgeSumAggregatorSparse_48430051230125
// MI455X (gfx1250) — compile-verified
//
#include <hip/hip_runtime.h>
#include <stdint.h>

// ---------------------------------------------------------------------------
// EdgeSumAggregatorSparse: out[b, tgt[e], :] += msgs[b, e, :]
//   B=2, E=500000, D=128, N=10000  (f32)
// Memory-bound scatter-add (AI ~ 0.25 flop/byte). Strategy:
//   - double-buffered async global->LDS staging (gfx1250 ASYNCcnt path)
//   - one wave32 per 512B row; 4 consecutive f32 global atomics per lane
// ---------------------------------------------------------------------------

#define AS1 __attribute__((address_space(1)))
#define AS3 __attribute__((address_space(3)))

typedef int v4i __attribute__((vector_size(16)));   // 128-bit payload type

constexpr int kB = 2;
constexpr int kE = 500000;
constexpr int kD = 128;
constexpr int kN = 10000;
constexpr int kR = kB * kE;              // 1,000,000 flattened rows
constexpr int kTILE = 64;                // rows staged per buffer (32 KB)
constexpr int kNTILES = kR / kTILE;      // 15625 (exact)
constexpr int kTHREADS = 256;            // 8 waves
constexpr int kROWS_PER_WAVE = kTILE / (kTHREADS / 32);   // 8

// ---- gfx1250 async copy: 16B per lane, global -> LDS ----------------------
__device__ __forceinline__ void async_copy_b128(const float* gsrc, float* ldst) {
#if __has_builtin(__builtin_amdgcn_global_load_async_to_lds_b128)
  __builtin_amdgcn_global_load_async_to_lds_b128(
      (AS1 v4i*)gsrc, (AS3 v4i*)ldst, 0, 0);
#else
  unsigned lds_off = (unsigned)(uint64_t)(AS3 char*)ldst;   // 32-bit LDS byte addr
  uint64_t gaddr   = (uint64_t)gsrc;
  asm volatile("global_load_async_to_lds_b128 %0, %1, off"
               :: "v"(lds_off), "v"(gaddr)
               : "memory");
#endif
}

__device__ __forceinline__ void wait_async_zero() {
#if __has_builtin(__builtin_amdgcn_s_wait_asynccnt)
  __builtin_amdgcn_s_wait_asynccnt(0);
#else
  asm volatile("s_wait_asynccnt 0" ::: "memory");
#endif
}

__device__ __forceinline__ void atomic_add_f32(float* p, float v) {
  // relaxed, device scope: should lower to a no-return f32 global atomic add
  // resolved in the L2 atomic units (output block fits in 192MB L2)
  __hip_atomic_fetch_add(p, v, __ATOMIC_RELAXED, __HIP_MEMORY_SCOPE_AGENT);
}

// ---------------------------------------------------------------------------
__global__ void __launch_bounds__(kTHREADS)
edge_scatter_kernel(const float* __restrict__ msgs,
                    const int* __restrict__ tgt,
                    float* __restrict__ out) {
  __shared__ __align__(16) float stage[2][kTILE][kD];   // 2 x 32 KB

  const int tid  = threadIdx.x;
  const int wave = tid >> 5;
  const int lane = tid & 31;

  // Stage one 32KB tile (64 contiguous rows of msgs) into stage[buf].
  // 256 threads x 8 x b128 = 32 KB, fully linear, 8 async ops/wave in flight.
  auto issue = [&](int tile, int buf) {
    const float* gsrc = msgs + (size_t)tile * (kTILE * kD);
    float* lbase = &stage[buf][0][0];
#pragma unroll
    for (int j = 0; j < 8; ++j) {
      const int off = j * (kTHREADS * 4) + tid * 4;     // float offset
      async_copy_b128(gsrc + off, lbase + off);
    }
  };

  // Drain tile from LDS: wave w owns rows [8w, 8w+8); lane owns 4 floats.
  auto consume = [&](int tile, int buf) {
#pragma unroll
    for (int i = 0; i < kROWS_PER_WAVE; ++i) {
      const int row = wave * kROWS_PER_WAVE + i;
      const int r   = tile * kTILE + row;               // flattened (b,e)
      const int b   = r / kE;
      const int e   = r - b * kE;
      int t = tgt[e];                                   // same addr across lanes
      t = __builtin_amdgcn_readfirstlane(t);            // force SGPR
      const float4 v =
          *reinterpret_cast<const float4*>(&stage[buf][row][lane * 4]);
      float* dst = out + ((size_t)b * kN + (size_t)t) * kD + lane * 4;
      atomic_add_f32(dst + 0, v.x);
      atomic_add_f32(dst + 1, v.y);
      atomic_add_f32(dst + 2, v.z);
      atomic_add_f32(dst + 3, v.w);
    }
  };

  int tile = blockIdx.x;
  int buf  = 0;
  if (tile < kNTILES) issue(tile, 0);
  wait_async_zero();
  __syncthreads();

  for (; tile < kNTILES; tile += gridDim.x) {
    const int nxt = tile + gridDim.x;
    if (nxt < kNTILES) issue(nxt, buf ^ 1);             // prefetch next tile
    consume(tile, buf);                                 // drain current tile
    wait_async_zero();                                  // next tile resident
    __syncthreads();                                    // visible to all waves
    buf ^= 1;
  }
}

// ---------------------------------------------------------------------------
__global__ void zero_f32x4_kernel(float* __restrict__ out, int n4) {
  const int i = blockIdx.x * blockDim.x + threadIdx.x;
  if (i < n4) reinterpret_cast<float4*>(out)[i] = float4{0.f, 0.f, 0.f, 0.f};
}

// ---------------------------------------------------------------------------
extern "C" void kernel_launch(void* const* d_in, const int* in_sizes, int n_in,
                              void* d_out, int out_size, void* d_ws, size_t ws_size,
                              hipStream_t stream) {
  const float* msgs = (const float*)d_in[0];   // [B,E,D] f32
  const int*   tgt  = (const int*)d_in[1];     // [E] i32
  float*       out  = (float*)d_out;           // [B,N,D] f32
  (void)in_sizes; (void)n_in; (void)d_ws; (void)ws_size;

  // out_size = B*N*D = 2,560,000 (divisible by 4)
  const int n4 = out_size / 4;
  zero_f32x4_kernel<<<(n4 + 255) / 256, 256, 0, stream>>>(out, n4);

  const int nblocks = (kNTILES < 4096) ? kNTILES : 4096;
  edge_scatter_kernel<<<nblocks, kTHREADS, 0, stream>>>(msgs, tgt, out);
}